// RGCNEncoder_71803263255134
// MI455X (gfx1250) — compile-verified
//
#include <hip/hip_runtime.h>

#define RELS 8

typedef __attribute__((ext_vector_type(2))) float v2f;
typedef __attribute__((ext_vector_type(8))) float v8f;

// ---------------------------------------------------------------------------
// Edge-count kernel: cnt[dst*R + etype] += 1  (identical for both layers)
// ---------------------------------------------------------------------------
__global__ void rgcn_count_kernel(const int* __restrict__ dst,
                                  const int* __restrict__ et,
                                  float* __restrict__ cnt, int E) {
  int e = blockIdx.x * blockDim.x + threadIdx.x;
  if (e < E) {
    unsafeAtomicAdd(&cnt[(size_t)dst[e] * RELS + (size_t)et[e]], 1.0f);
  }
}

// ---------------------------------------------------------------------------
// Scatter-sum of raw features: sums[(dst*R + etype), :] += feat[src, :]
// One thread per (edge, 4-float chunk); float4 gather, 4x f32 hardware atomics.
// ---------------------------------------------------------------------------
__global__ void rgcn_scatter_kernel(const float* __restrict__ feat,
                                    const int* __restrict__ src,
                                    const int* __restrict__ dst,
                                    const int* __restrict__ et,
                                    float* __restrict__ sums,
                                    int E, int K) {
  const int chunks = K >> 2;
  long long tid = (long long)blockIdx.x * blockDim.x + threadIdx.x;
  long long total = (long long)E * chunks;
  if (tid >= total) return;
  int e = (int)(tid / chunks);
  int c = (int)(tid % chunks) << 2;
  int s = src[e], d = dst[e], r = et[e];
  const float4 v = *reinterpret_cast<const float4*>(feat + (size_t)s * K + c);
  float* p = sums + ((size_t)d * RELS + r) * K + c;
  unsafeAtomicAdd(p + 0, v.x);
  unsafeAtomicAdd(p + 1, v.y);
  unsafeAtomicAdd(p + 2, v.z);
  unsafeAtomicAdd(p + 3, v.w);
}

// ---------------------------------------------------------------------------
// Fused RGCN GEMM (fp32 WMMA):
//   out[n, m] = bias[m] + Sum_r mean(n,r,:)·W[r,:,m] + x[n,:]·Wroot[:,m]
// Effective K = (R+1)*K. One wave -> 16x64 output tile via 4x
// V_WMMA_F32_16X16X4_F32 accumulators; 4 waves/block -> 64x64 tile.
// mean = sums * 1/max(cnt,1) folded into the A-operand load.
// All 8 B operands are fetched into distinct registers before the 4 WMMAs
// so the compiler can batch the loads (single wait) and issue the WMMAs
// back-to-back.
// ---------------------------------------------------------------------------
__device__ __forceinline__ v8f wmma_f32(v2f a, v2f b, v8f c) {
  return __builtin_amdgcn_wmma_f32_16x16x4_f32(false, a, false, b,
                                               (short)0, c, false, false);
}

__global__ void __launch_bounds__(128)
rgcn_gemm_kernel(const float* __restrict__ sums,
                 const float* __restrict__ cnt,
                 const float* __restrict__ xin,
                 const float* __restrict__ W,      // [R, K, M]
                 const float* __restrict__ Wroot,  // [K, M]
                 const float* __restrict__ bias,   // [M]
                 float* __restrict__ out,          // [N, M]
                 int N, int K, int M, int relu) {
  const int wave = threadIdx.x >> 5;
  const int lane = threadIdx.x & 31;
  const int half = lane >> 4;   // which 16-lane half of the wave
  const int lr   = lane & 15;
  const int rowBase = blockIdx.x * 64 + wave * 16;
  const int colBase = blockIdx.y * 64;

  // Clamp A-operand row so EXEC stays all-ones through the WMMA loop.
  int rowA = rowBase + lr;
  if (rowA > N - 1) rowA = N - 1;

  v8f acc[4] = {};

  for (int r = 0; r <= RELS; ++r) {
    const float* Aptr;
    const float* Bptr;
    float scale;
    if (r < RELS) {
      Aptr  = sums + ((size_t)rowA * RELS + r) * K;
      Bptr  = W + (size_t)r * K * M;
      scale = 1.0f / fmaxf(cnt[(size_t)rowA * RELS + r], 1.0f);
    } else {
      Aptr  = xin + (size_t)rowA * K;
      Bptr  = Wroot;
      scale = 1.0f;
    }
    #pragma unroll 4
    for (int k = 0; k < K; k += 4) {
      const int ka = k + 2 * half;  // ISA f32 A layout: v0=K0/K2, v1=K1/K3
      const float2 av = *reinterpret_cast<const float2*>(Aptr + ka);
      const float* Brow0 = Bptr + (size_t)ka * M + colBase + lr;
      const float* Brow1 = Brow0 + M;

      // Stage all B operands first (8 independent loads -> one wait), ...
      v2f b[4];
      #pragma unroll
      for (int t = 0; t < 4; ++t) {
        b[t].x = Brow0[t * 16];
        b[t].y = Brow1[t * 16];
      }
      v2f a;
      a.x = av.x * scale;
      a.y = av.y * scale;
      // ... then issue the 4 WMMAs back-to-back.
      #pragma unroll
      for (int t = 0; t < 4; ++t) {
        acc[t] = wmma_f32(a, b[t], acc[t]);
      }
    }
  }

  // C/D layout: VGPR v, lanes 0-15 -> row v; lanes 16-31 -> row v+8.
  #pragma unroll
  for (int t = 0; t < 4; ++t) {
    const int col = colBase + t * 16 + lr;
    const float bv = bias[col];
    #pragma unroll
    for (int v = 0; v < 8; ++v) {
      const int rr = rowBase + v + 8 * half;
      if (rr < N) {
        float val = acc[t][v] + bv;
        if (relu) val = fmaxf(val, 0.0f);
        out[(size_t)rr * M + col] = val;
      }
    }
  }
}

// ---------------------------------------------------------------------------
// Host-side orchestration (all on `stream`; graph-capture safe).
// ---------------------------------------------------------------------------
extern "C" void kernel_launch(void* const* d_in, const int* in_sizes, int n_in,
                              void* d_out, int out_size, void* d_ws, size_t ws_size,
                              hipStream_t stream) {
  const float* x     = (const float*)d_in[0];
  const int*   ei    = (const int*)d_in[1];   // [2, E]
  const int*   et    = (const int*)d_in[2];   // [E]
  const float* W1    = (const float*)d_in[3]; // [R, di, dh]
  const float* root1 = (const float*)d_in[4]; // [di, dh]
  const float* b1    = (const float*)d_in[5]; // [dh]
  const float* W2    = (const float*)d_in[6]; // [R, dh, do]
  const float* root2 = (const float*)d_in[7]; // [dh, do]
  const float* b2    = (const float*)d_in[8]; // [do]
  float*       out   = (float*)d_out;

  const int E  = in_sizes[2];
  const int dh = in_sizes[5];
  const int di = in_sizes[4] / dh;
  const int dO = in_sizes[8];
  const int N  = in_sizes[0] / di;
  const int* src = ei;
  const int* dst = ei + E;

  // Workspace layout (floats): cnt[N*R] | h[N*dh] | sums[N*R*max(di,dh)]
  float* cnt  = (float*)d_ws;
  float* h    = cnt + (size_t)N * RELS;
  float* sums = h + (size_t)N * dh;

  // ---- Edge counts (shared by both layers) ----
  hipMemsetAsync(cnt, 0, (size_t)N * RELS * sizeof(float), stream);
  rgcn_count_kernel<<<(E + 255) / 256, 256, 0, stream>>>(dst, et, cnt, E);

  // ---- Layer 1: scatter x, fused GEMM -> h (with ReLU) ----
  hipMemsetAsync(sums, 0, (size_t)N * RELS * di * sizeof(float), stream);
  {
    long long total = (long long)E * (di / 4);
    rgcn_scatter_kernel<<<(int)((total + 255) / 256), 256, 0, stream>>>(
        x, src, dst, et, sums, E, di);
    dim3 grid((N + 63) / 64, dh / 64);
    rgcn_gemm_kernel<<<grid, 128, 0, stream>>>(
        sums, cnt, x, W1, root1, b1, h, N, di, dh, /*relu=*/1);
  }

  // ---- Layer 2: scatter h, fused GEMM -> out ----
  hipMemsetAsync(sums, 0, (size_t)N * RELS * dh * sizeof(float), stream);
  {
    long long total = (long long)E * (dh / 4);
    rgcn_scatter_kernel<<<(int)((total + 255) / 256), 256, 0, stream>>>(
        h, src, dst, et, sums, E, dh);
    dim3 grid((N + 63) / 64, dO / 64);
    rgcn_gemm_kernel<<<grid, 128, 0, stream>>>(
        sums, cnt, h, W2, root2, b2, out, N, dh, dO, /*relu=*/0);
  }
}